// DistLoss_10359461118287
// MI455X (gfx1250) — compile-verified
//
#include <hip/hip_runtime.h>
#include <hip/hip_bf16.h>
#include <stdint.h>

#define BS   8
#define LQ   256
#define CH   3
#define HH   192
#define WW   192
#define HW   (HH*WW)          // 36864
#define KTOP 16
#define WAVES 8
#define TILES_PER_WAVE (HW/16/WAVES)   // 288

typedef __attribute__((ext_vector_type(2))) float v2f;
typedef __attribute__((ext_vector_type(8))) float v8f;

// ---------------------------------------------------------------------------
// tournament-tree argmax over 16 registers (depth 4 instead of serial 15)
// ---------------------------------------------------------------------------
__device__ __forceinline__ void recompute_max16(const float (&td)[KTOP],
                                                float& tmax, int& tmaxi) {
    float m8[8]; int i8[8];
    #pragma unroll
    for (int s = 0; s < 8; ++s) {
        bool g = td[2*s+1] > td[2*s];
        m8[s] = g ? td[2*s+1] : td[2*s];
        i8[s] = g ? 2*s+1 : 2*s;
    }
    float m4[4]; int i4[4];
    #pragma unroll
    for (int s = 0; s < 4; ++s) {
        bool g = m8[2*s+1] > m8[2*s];
        m4[s] = g ? m8[2*s+1] : m8[2*s];
        i4[s] = g ? i8[2*s+1] : i8[2*s];
    }
    float m2[2]; int i2[2];
    #pragma unroll
    for (int s = 0; s < 2; ++s) {
        bool g = m4[2*s+1] > m4[2*s];
        m2[s] = g ? m4[2*s+1] : m4[2*s];
        i2[s] = g ? i4[2*s+1] : i4[2*s];
    }
    bool g = m2[1] > m2[0];
    tmax  = g ? m2[1] : m2[0];
    tmaxi = g ? i2[1] : i2[0];
}

__device__ __forceinline__ void insert16(float (&td)[KTOP], int (&ti)[KTOP],
                                         float& tmax, int& tmaxi,
                                         float dv, int idx) {
    #pragma unroll
    for (int s = 0; s < KTOP; ++s)
        if (s == tmaxi) { td[s] = dv; ti[s] = idx; }
    recompute_max16(td, tmax, tmaxi);
}

// ---------------------------------------------------------------------------
// Kernel 1a: pack per-pixel B data  {-2r, rsq}  -> Bpk[b][hw][4]
// ---------------------------------------------------------------------------
__global__ void prep_B_kernel(const float* __restrict__ ref, float* __restrict__ Bpk) {
    int i = blockIdx.x * blockDim.x + threadIdx.x;
    if (i >= BS * HW) return;
    int b = i / HW, hw = i % HW;
    const float* rb = ref + (size_t)b * CH * HW + hw;
    float r0 = rb[0], r1 = rb[HW], r2 = rb[2 * HW];
    float4 o;
    o.x = -2.f * r0; o.y = -2.f * r1; o.z = -2.f * r2;
    o.w = r0 * r0 + r1 * r1 + r2 * r2;
    ((float4*)Bpk)[i] = o;
}

// ---------------------------------------------------------------------------
// Kernel 1b: gather pooled colors (scrambled flat indexing of the reference),
// pack A data {p0,p1,p2,1} and psq.
// pooled[b][l] <- pooled_flat[i = l*8 + b], grid source = pos[i>>8][i&255]
// ---------------------------------------------------------------------------
__global__ void prep_A_kernel(const float* __restrict__ pred, const float* __restrict__ ref,
                              float* __restrict__ Apk, float* __restrict__ Cps) {
    int i = blockIdx.x * blockDim.x + threadIdx.x;   // pooled_flat index
    if (i >= BS * LQ) return;
    int b  = i & (BS - 1);
    int l  = i >> 3;
    int bp = i >> 8;          // pos source batch
    int lp = i & (LQ - 1);    // pos source row
    float gx = pred[(size_t)bp * LQ * 8 + lp * 8 + 0];
    float gy = pred[(size_t)bp * LQ * 8 + lp * 8 + 1];
    float fx = rintf(gx * (float)WW - 0.5f);   // round-half-even like jnp.round
    float fy = rintf(gy * (float)HH - 0.5f);
    int ix = (int)fx, iy = (int)fy;
    float inb = (ix >= 0 && ix < WW && iy >= 0 && iy < HH) ? 1.f : 0.f;
    int ixc = min(max(ix, 0), WW - 1);
    int iyc = min(max(iy, 0), HH - 1);
    const float* rb = ref + (size_t)b * CH * HW + iyc * WW + ixc;
    float p0 = rb[0] * inb, p1 = rb[HW] * inb, p2 = rb[2 * HW] * inb;
    float4 a; a.x = p0; a.y = p1; a.z = p2; a.w = 1.f;
    ((float4*)Apk)[b * LQ + l] = a;
    Cps[b * LQ + l] = p0 * p0 + p1 * p1 + p2 * p2;
}

// ---------------------------------------------------------------------------
// Kernel 2: fused WMMA distance + per-query top-16, software-pipelined.
// Block = 256 thr (8 waves); 128 blocks = (batch, 16-query tile).
// D[pixel,query] = A'(pixel tile 16x4) x B'(query tile 4x16) + psq[q].
// Lane (half=lane>>4, q=lane&15) owns query q, pixels base+half*8+v.
// ---------------------------------------------------------------------------
__global__ __launch_bounds__(256) void dist_topk_kernel(
        const float* __restrict__ Apk, const float* __restrict__ Cps,
        const float* __restrict__ Bpk, float* __restrict__ Tgt) {

    __shared__ float mergeD[16][16][KTOP];   // [query][slot][entry]
    __shared__ int   mergeI[16][16][KTOP];

    const int b    = blockIdx.x >> 4;
    const int lt   = blockIdx.x & 15;
    const int tid  = threadIdx.x;
    const int w    = tid >> 5;
    const int lane = tid & 31;
    const int half = lane >> 4;   // 0: K={0,1}/pixels 0-7 ; 1: K={2,3}/pixels 8-15
    const int q    = lane & 15;   // query column owned by this lane

    const float* Ab = Apk + ((size_t)b * LQ + lt * 16) * 4;   // query data
    const float* Cb = Cps + (size_t)b * LQ + lt * 16;
    const float* Bb = Bpk + (size_t)b * HW * 4;               // pixel data
    const float* Bl = Bb + (size_t)q * 4 + half * 2;          // lane's B stream

    // B operand (4x16 query matrix, loop invariant): VGPR0=K{0,2}, VGPR1=K{1,3}
    v2f bq;
    { const float* p = Ab + q * 4 + half * 2; bq.x = p[0]; bq.y = p[1]; }

    // C operand: psq of this lane's query, broadcast over the 8 pixel rows
    v8f c;
    { float psq = Cb[q];
      #pragma unroll
      for (int v = 0; v < 8; ++v) c[v] = psq; }

    // in-register top-16 (unsorted, tracked max)
    float td[KTOP]; int ti[KTOP];
    #pragma unroll
    for (int s = 0; s < KTOP; ++s) { td[s] = 3.0e38f; ti[s] = 0; }
    float tmax = 3.0e38f; int tmaxi = KTOP - 1;

    const int t0 = w * TILES_PER_WAVE;

    // --- software pipeline: prime tile 0's A operand ---------------------
    v2f ap_cur;
    { const float* p = Bl + (size_t)t0 * 64;           // (t0*16)*4 floats
      ap_cur.x = p[0]; ap_cur.y = p[1]; }

    for (int t = 0; t < TILES_PER_WAVE; ++t) {
        const int base = (t0 + t) * 16;

        // issue next tile's load before consuming the current one
        const int tn = (t + 1 < TILES_PER_WAVE) ? (t + 1) : t;   // branchless clamp
        v2f ap_next;
        { const float* p = Bl + (size_t)(t0 + tn) * 64;
          ap_next.x = p[0]; ap_next.y = p[1]; }
        __builtin_prefetch((const void*)(Bb + (size_t)(base + 256) * 4), 0, 1);

        // D[pixel, query] = ap x bq + psq   (exact squared color distance)
        v8f d = __builtin_amdgcn_wmma_f32_16x16x4_f32(
                    false, ap_cur, false, bq, (short)0, c, false, false);

        #pragma unroll
        for (int v = 0; v < 8; ++v) {
            float dv = d[v];
            if (dv < tmax)
                insert16(td, ti, tmax, tmaxi, dv, base + half * 8 + v);
        }
        ap_cur = ap_next;
    }

    // dump partial lists: 16 slots per query (8 waves x 2 halves)
    const int slot = w * 2 + half;
    #pragma unroll
    for (int s = 0; s < KTOP; ++s) {
        mergeD[q][slot][s] = td[s];
        mergeI[q][slot][s] = ti[s];
    }
    __syncthreads();

    // wave 0, lanes 0-15: merge 256 candidates per query, rank-sort, emit tgt
    if (tid < 16) {
        const int mq = tid;
        float fd[KTOP]; int fi[KTOP];
        #pragma unroll
        for (int s = 0; s < KTOP; ++s) { fd[s] = 3.0e38f; fi[s] = 0x7fffffff; }
        float fmax = 3.0e38f; int fmaxi = KTOP - 1;
        for (int sl = 0; sl < 16; ++sl) {
            #pragma unroll
            for (int e = 0; e < KTOP; ++e) {
                float dv = mergeD[mq][sl][e];
                int   iv = mergeI[mq][sl][e];
                if (dv < fmax)
                    insert16(fd, fi, fmax, fmaxi, dv, iv);
            }
        }
        // rank-sort (ascending distance, ties -> lower pixel index, like top_k)
        float* out = Tgt + ((size_t)(b * LQ + lt * 16 + mq) * KTOP) * 2;
        #pragma unroll
        for (int i = 0; i < KTOP; ++i) {
            int rank = 0;
            #pragma unroll
            for (int j = 0; j < KTOP; ++j)
                rank += (fd[j] < fd[i]) || (fd[j] == fd[i] && fi[j] < fi[i]);
            float tx = (float)(fi[i] % WW) * (1.f / (float)WW);
            float ty = (float)(fi[i] / WW) * (1.f / (float)WW);
            out[2 * rank + 0] = tx;
            out[2 * rank + 1] = ty;
        }
    }
}

// ---------------------------------------------------------------------------
// Kernel 3: roll targets by 1 along L, pick closest of 16, mean squared loss.
// Single workgroup -> writes the scalar directly (d_out is poisoned).
// ---------------------------------------------------------------------------
__global__ __launch_bounds__(256) void final_loss_kernel(
        const float* __restrict__ pred, const float* __restrict__ Tgt,
        float* __restrict__ out) {
    __shared__ float red[256];
    const int tid = threadIdx.x;
    float acc = 0.f;
    for (int g = tid; g < BS * LQ; g += 256) {
        int b = g >> 8, l = g & 255;
        if (l == 0) continue;                       // loss uses l >= 1 only
        float px = pred[(size_t)b * LQ * 8 + l * 8 + 0];
        float py = pred[(size_t)b * LQ * 8 + l * 8 + 1];
        const float* tg = Tgt + ((size_t)(b * LQ + (l - 1)) * KTOP) * 2;  // rolled
        float best = 3.4e38f, fx = 0.f, fy = 0.f;
        #pragma unroll
        for (int k = 0; k < KTOP; ++k) {
            float tx = tg[2 * k], ty = tg[2 * k + 1];
            float dd = (px - tx) * (px - tx) + (py - ty) * (py - ty);
            if (dd < best) { best = dd; fx = tx; fy = ty; }   // ties -> lowest k
        }
        acc += (px - fx) * (px - fx) + (py - fy) * (py - fy);
    }
    red[tid] = acc;
    __syncthreads();
    for (int s = 128; s > 0; s >>= 1) {
        if (tid < s) red[tid] += red[tid + s];
        __syncthreads();
    }
    if (tid == 0) out[0] = red[0] / (float)(BS * (LQ - 1));
}

// ---------------------------------------------------------------------------
extern "C" void kernel_launch(void* const* d_in, const int* in_sizes, int n_in,
                              void* d_out, int out_size, void* d_ws, size_t ws_size,
                              hipStream_t stream) {
    const float* pred = (const float*)d_in[0];   // (8,256,8)
    const float* ref  = (const float*)d_in[1];   // (8,3,192,192)

    float* ws  = (float*)d_ws;
    float* Bpk = ws;                                   // 8*36864*4 floats (4.5 MB)
    float* Apk = Bpk + (size_t)BS * HW * 4;            // 8*256*4
    float* Cps = Apk + (size_t)BS * LQ * 4;            // 8*256
    float* Tgt = Cps + (size_t)BS * LQ;                // 8*256*16*2

    prep_B_kernel<<<(BS * HW + 255) / 256, 256, 0, stream>>>(ref, Bpk);
    prep_A_kernel<<<(BS * LQ + 255) / 256, 256, 0, stream>>>(pred, ref, Apk, Cps);
    dist_topk_kernel<<<BS * 16, 256, 0, stream>>>(Apk, Cps, Bpk, Tgt);
    final_loss_kernel<<<1, 256, 0, stream>>>(pred, Tgt, (float*)d_out);
}